// ModelNew_85710367359296
// MI455X (gfx1250) — compile-verified
//
#include <hip/hip_runtime.h>
#include <hip/hip_bf16.h>

// ---- problem constants (from reference) ----
#define BATCH   4
#define SEQ     4096
#define HEADS   16
#define PDIM    64
#define NDIM    64
#define TSLICE  256                 // timesteps per GEMM block (SEQ/TSLICE = 16 k-slices)
#define TS      16                  // timesteps per LDS slab
#define NSLAB   (TSLICE / TS)       // 16 slabs per block
#define LROW    72                  // padded LDS row stride (floats); 72 % 64 = 8 -> rows t and t+2
                                    // are 16 banks apart -> half-waves hit disjoint banks

typedef __attribute__((ext_vector_type(2))) float v2f;
typedef __attribute__((ext_vector_type(4))) float v4f;
typedef __attribute__((ext_vector_type(8))) float v8f;
typedef __attribute__((ext_vector_type(4))) int   v4i;

// exact parameter types of __builtin_amdgcn_global_load_async_to_lds_b128
// (from the round-2 diagnostic: arg0 is int4 in AS(1); dest is int4 in AS(3))
typedef __attribute__((address_space(1))) v4i* gv4i_p;
typedef __attribute__((address_space(3))) v4i* lv4i_p;

#if defined(__gfx1250__) && \
    __has_builtin(__builtin_amdgcn_global_load_async_to_lds_b128) && \
    __has_builtin(__builtin_amdgcn_s_wait_asynccnt)
#define USE_ASYNC_LDS 1
#else
#define USE_ASYNC_LDS 0
#endif

// ------------------------------------------------------------------
// Kernel 0: zero the output (harness poisons d_out; we accumulate).
// ------------------------------------------------------------------
__global__ void ssd_zero_out(float* __restrict__ out, int n) {
    int i = blockIdx.x * blockDim.x + threadIdx.x;
    if (i < n) out[i] = 0.0f;
}

// ------------------------------------------------------------------
// Kernel 1: per-(b,h) weights  W[bh, t] = exp(total - inclusive_cumsum(t)).
// One block (256 threads) per (b,h). A is (b, s, h) -> stride t = HEADS.
// ------------------------------------------------------------------
__global__ void ssd_scan_weights(const float* __restrict__ A, float* __restrict__ W) {
    const int bh = blockIdx.x;                 // 0..63
    const int b  = bh / HEADS;
    const int h  = bh % HEADS;
    const float* Ab = A + (size_t)b * SEQ * HEADS + h;   // Ab[t*HEADS]

    __shared__ float sdata[256];
    __shared__ float s_total;
    __shared__ float s_carry;
    const int tid = threadIdx.x;

    // ---- pass 1: total sum over 4096 elements ----
    float tsum = 0.0f;
    for (int t = tid; t < SEQ; t += 256) tsum += Ab[(size_t)t * HEADS];
    sdata[tid] = tsum;
    __syncthreads();
    for (int off = 128; off > 0; off >>= 1) {
        if (tid < off) sdata[tid] += sdata[tid + off];
        __syncthreads();
    }
    if (tid == 0) { s_total = sdata[0]; s_carry = 0.0f; }
    __syncthreads();
    const float total = s_total;

    // ---- pass 2: tiled inclusive scan, emit exp(total - cumsum) ----
    float* Wp = W + (size_t)bh * SEQ;
    for (int base = 0; base < SEQ; base += 256) {
        float v = Ab[(size_t)(base + tid) * HEADS];
        sdata[tid] = v;
        __syncthreads();
        // Hillis-Steele: read-phase / barrier / write-phase per step (race-free)
        for (int off = 1; off < 256; off <<= 1) {
            float add = (tid >= off) ? sdata[tid - off] : 0.0f;
            __syncthreads();
            sdata[tid] += add;
            __syncthreads();
        }
        float incl = sdata[tid] + s_carry;
        Wp[base + tid] = __expf(total - incl);
        __syncthreads();
        if (tid == 255) s_carry = incl;
        __syncthreads();
    }
}

// ------------------------------------------------------------------
// Kernel 2: decay-weighted outer-product GEMM via V_WMMA_F32_16X16X4_F32,
// with double-buffered LDS staging (async-to-LDS when available).
// Grid: (kslice=16, h=16, b=4); block: 256 threads = 8 wave32.
// Wave w: p-tile pi = w>>1, n-tiles {2*(w&1), 2*(w&1)+1}.
// A (16x4 f32): lanes 0-15 -> K=0,1 ; lanes 16-31 -> K=2,3 ; M = lane&15.
// D (16x16 f32): vgpr r -> M = (lane>>4)*8 + r, N = lane&15.
// ------------------------------------------------------------------
__global__ void __launch_bounds__(256)
ssd_wmma_gemm(const float* __restrict__ X, const float* __restrict__ Bm,
              const float* __restrict__ W, float* __restrict__ out) {
    __shared__ alignas(16) float lds_x[2][TS * LROW];
    __shared__ alignas(16) float lds_b[2][TS * LROW];
    __shared__ float lds_w[2][TS];

    const int ks = blockIdx.x;        // k-slice 0..15
    const int h  = blockIdx.y;        // 0..15
    const int b  = blockIdx.z;        // 0..3

    const int tid  = threadIdx.x;
    const int wave = tid >> 5;             // 0..7 (wave32)
    const int lane = tid & 31;
    const int pi = wave >> 1;              // p-tile 0..3
    const int nj = (wave & 1) * 2;         // n-tiles nj, nj+1
    const int m  = lane & 15;              // row/col within 16-tile
    const int kb = (lane >> 4) << 1;       // 0 or 2: K-pair this half-wave holds

    const size_t strideT = (size_t)HEADS * PDIM;   // elements per timestep
    const size_t gbase   = ((size_t)b * SEQ + (size_t)ks * TSLICE) * strideT
                         + (size_t)h * PDIM;
    const float* Xp = X  + gbase;
    const float* Bp = Bm + gbase;
    const float* Wp = W + ((size_t)(b * HEADS + h)) * SEQ + (size_t)ks * TSLICE;

    // staging coords: 256 threads cover one slab (16 t x 64 floats) per matrix,
    // one b128 per thread per matrix, fully coalesced (16 lanes -> 256B row).
    const int st = tid >> 4;            // timestep within slab, 0..15
    const int sp = (tid & 15) << 2;     // float4 column, 0..60

    auto stage = [&](int slab, int buf) {
        const float* gx = Xp + (size_t)(slab * TS + st) * strideT + sp;
        const float* gb = Bp + (size_t)(slab * TS + st) * strideT + sp;
        const int li = st * LROW + sp;
#if USE_ASYNC_LDS
        __builtin_amdgcn_global_load_async_to_lds_b128(
            (gv4i_p)gx, (lv4i_p)&lds_x[buf][li], 0, 0);
        __builtin_amdgcn_global_load_async_to_lds_b128(
            (gv4i_p)gb, (lv4i_p)&lds_b[buf][li], 0, 0);
#else
        *(v4f*)&lds_x[buf][li] = *(const v4f*)gx;
        *(v4f*)&lds_b[buf][li] = *(const v4f*)gb;
#endif
        if (tid < TS) lds_w[buf][tid] = Wp[slab * TS + tid];
    };

    v8f acc0 = {};
    v8f acc1 = {};

    stage(0, 0);
    for (int s = 0; s < NSLAB; ++s) {
        const int buf = s & 1;
#if USE_ASYNC_LDS
        __builtin_amdgcn_s_wait_asynccnt(0);   // my async copies into `buf` done
#endif
        __syncthreads();                       // everyone's staging of `buf` visible;
                                               // also fences reads of buf^1 from last iter
        if (s + 1 < NSLAB) stage(s + 1, buf ^ 1);

        const float* xs = lds_x[buf];
        const float* bs = lds_b[buf];
        const float* ws = lds_w[buf];
#pragma unroll
        for (int kk = 0; kk < TS; kk += 4) {
            const int t0 = kk + kb;
            const float w0 = ws[t0];
            const float w1 = ws[t0 + 1];

            v2f a;
            a.x = w0 * xs[t0 * LROW + pi * 16 + m];
            a.y = w1 * xs[(t0 + 1) * LROW + pi * 16 + m];

            v2f b0, b1;
            b0.x = bs[t0 * LROW + (nj + 0) * 16 + m];
            b0.y = bs[(t0 + 1) * LROW + (nj + 0) * 16 + m];
            b1.x = bs[t0 * LROW + (nj + 1) * 16 + m];
            b1.y = bs[(t0 + 1) * LROW + (nj + 1) * 16 + m];

            acc0 = __builtin_amdgcn_wmma_f32_16x16x4_f32(
                false, a, false, b0, (short)0, acc0, false, false);
            acc1 = __builtin_amdgcn_wmma_f32_16x16x4_f32(
                false, a, false, b1, (short)0, acc1, false, false);
        }
    }

    // Accumulate partial 64x64 tile into out[b,h,:,:] (16 k-slices race -> atomics).
    float* outp = out + ((size_t)(b * HEADS + h)) * PDIM * NDIM;
    const int n_lane = lane & 15;
    const int m_base = (lane >> 4) * 8;
#pragma unroll
    for (int r = 0; r < 8; ++r) {
        const int mm = pi * 16 + m_base + r;
        atomicAdd(&outp[(size_t)mm * NDIM + (nj + 0) * 16 + n_lane], acc0[r]);
        atomicAdd(&outp[(size_t)mm * NDIM + (nj + 1) * 16 + n_lane], acc1[r]);
    }
}

// ------------------------------------------------------------------
// Launch wrapper
// ------------------------------------------------------------------
extern "C" void kernel_launch(void* const* d_in, const int* in_sizes, int n_in,
                              void* d_out, int out_size, void* d_ws, size_t ws_size,
                              hipStream_t stream) {
    const float* X = (const float*)d_in[0];   // (4, 4096, 16, 64)
    const float* A = (const float*)d_in[1];   // (4, 4096, 16)
    const float* B = (const float*)d_in[2];   // (4, 4096, 16, 64)
    // d_in[3] (C) is dead code in the reference -> unused.
    float* out = (float*)d_out;               // (4, 16, 64, 64) f32
    float* W   = (float*)d_ws;                // 4*16*4096 floats = 1 MB scratch

    // 0) zero the accumulator output
    ssd_zero_out<<<(out_size + 255) / 256, 256, 0, stream>>>(out, out_size);
    // 1) per-(b,h) decay weights
    ssd_scan_weights<<<BATCH * HEADS, 256, 0, stream>>>(A, W);
    // 2) WMMA f32 GEMM, k-split over 16 slices for occupancy/bandwidth
    dim3 grid(SEQ / TSLICE, HEADS, BATCH);   // (16, 16, 4) = 1024 blocks
    ssd_wmma_gemm<<<grid, 256, 0, stream>>>(X, B, W, out);
}